// AttnDecoderRNN_42889543417942
// MI455X (gfx1250) — compile-verified
//
#include <hip/hip_runtime.h>
#include <hip/hip_bf16.h>
#include <math.h>

// ---------------------------------------------------------------------------
// Types for WMMA f32 16x16x4: A=16x4 f32 (2 VGPR/lane), B=4x16 f32 (2 VGPR),
// C/D=16x16 f32 (8 VGPR/lane).
// ---------------------------------------------------------------------------
typedef __attribute__((ext_vector_type(2))) float v2f;
typedef __attribute__((ext_vector_type(8))) float v8f;

// ---------------------------------------------------------------------------
// Wave/block reductions (wave32)
// ---------------------------------------------------------------------------
__device__ inline float wave_max(float v) {
    for (int o = 16; o > 0; o >>= 1) v = fmaxf(v, __shfl_xor(v, o, 32));
    return v;
}
__device__ inline float wave_sum(float v) {
    for (int o = 16; o > 0; o >>= 1) v += __shfl_xor(v, o, 32);
    return v;
}
__device__ inline float block_max(float v, float* sm) {
    const int lane = threadIdx.x & 31, wid = threadIdx.x >> 5;
    const int nw = (blockDim.x + 31) >> 5;
    v = wave_max(v);
    if (lane == 0) sm[wid] = v;
    __syncthreads();
    if (wid == 0) {
        float t = (threadIdx.x < nw) ? sm[threadIdx.x] : -INFINITY;
        t = wave_max(t);
        if (lane == 0) sm[0] = t;
    }
    __syncthreads();
    float r = sm[0];
    __syncthreads();
    return r;
}
__device__ inline float block_sum(float v, float* sm) {
    const int lane = threadIdx.x & 31, wid = threadIdx.x >> 5;
    const int nw = (blockDim.x + 31) >> 5;
    v = wave_sum(v);
    if (lane == 0) sm[wid] = v;
    __syncthreads();
    if (wid == 0) {
        float t = (threadIdx.x < nw) ? sm[threadIdx.x] : 0.0f;
        t = wave_sum(t);
        if (lane == 0) sm[0] = t;
    }
    __syncthreads();
    float r = sm[0];
    __syncthreads();
    return r;
}

// ---------------------------------------------------------------------------
// One GEMV reduction phase on V_WMMA_F32_16X16X4_F32.
// A layout (ISA 16x4 f32): lanes 0-15 hold rows M=0..15 with K=k,k+1 in
// VGPR0/1; lanes 16-31 hold K=k+2,k+3.  B mirrors with N striped on lanes;
// all N columns carry the same broadcast x values.
// ---------------------------------------------------------------------------
__device__ inline v8f wmma_gemv_phase(v8f acc, const float* __restrict__ wr,
                                      const float* xv, int K, int koff) {
    for (int k = 0; k < K; k += 32) {
        __builtin_prefetch(wr + k + 512, 0, 0);  // global_prefetch_b8
#pragma unroll
        for (int kk = 0; kk < 32; kk += 4) {
            const int kx = k + kk + koff;
            v2f a = *(const v2f*)(wr + kx);  // 16 weight rows, b64 per lane
            v2f b = *(const v2f*)(xv + kx);  // x broadcast from LDS
            acc = __builtin_amdgcn_wmma_f32_16x16x4_f32(
                false, a, false, b, (short)0, acc, false, false);
        }
    }
    return acc;
}

// y[r] = act( dot(W1[r,:],x1) + b1[r] [+ dot(W2[r,:],x2) + b2[r]] )
// One wave per 16-row tile; K1,K2 multiples of 32; K1+K2 <= 2048.
__global__ void __launch_bounds__(256)
wmma_gemv_kernel(const float* __restrict__ W1, const float* __restrict__ x1, int K1,
                 const float* __restrict__ W2, const float* __restrict__ x2, int K2,
                 const float* __restrict__ b1, const float* __restrict__ b2,
                 float* __restrict__ out, int R, int act) {
    __shared__ float sx[2048];
    const int tid = threadIdx.x;
    for (int i = tid; i < K1; i += blockDim.x) sx[i] = x1[i];
    if (W2)
        for (int i = tid; i < K2; i += blockDim.x) sx[K1 + i] = x2[i];
    __syncthreads();

    const int wave = tid >> 5;
    const int lane = tid & 31;
    const int tile = blockIdx.x * (blockDim.x >> 5) + wave;
    const int row0 = tile * 16;
    if (row0 >= R) return;  // wave-uniform

    int rowm = row0 + (lane & 15);
    if (rowm > R - 1) rowm = R - 1;          // tail tile: duplicate last row
    const int koff = (lane >> 4) << 1;       // 0 for lanes 0-15, 2 for 16-31

    v8f acc = {};
    acc = wmma_gemv_phase(acc, W1 + (size_t)rowm * K1, sx, K1, koff);
    if (W2)
        acc = wmma_gemv_phase(acc, W2 + (size_t)rowm * K2, sx + K1, K2, koff);

    // D column N=0: lane 0 holds M=0..7 in acc[0..7]; lane 16 holds M=8..15.
    if ((lane & 15) == 0) {
        const int mb = (lane >> 4) * 8;
#pragma unroll
        for (int q = 0; q < 8; ++q) {
            const int r = row0 + mb + q;
            if (r < R) {
                float y = acc[q];
                if (b1) y += b1[r];
                if (b2) y += b2[r];
                if (act) y = fmaxf(y, 0.0f);
                out[r] = y;
            }
        }
    }
}

// ---------------------------------------------------------------------------
// Stage 1: embedded = emb[token]; attn_in = [embedded | h0]; vecB[0:1024]=emb
// ---------------------------------------------------------------------------
__global__ void __launch_bounds__(256)
build_vecs_kernel(const int* __restrict__ tok, const float* __restrict__ emb,
                  const float* __restrict__ h, float* __restrict__ attn_in,
                  float* __restrict__ vecB) {
    const int j = blockIdx.x * blockDim.x + threadIdx.x;  // 0..2047
    const int t = tok[0];
    if (j < 1024) {
        const float e = emb[(size_t)t * 1024 + j];
        attn_in[j] = e;
        vecB[j] = e;
    } else if (j < 2048) {
        attn_in[j] = h[j - 1024];
    }
}

// ---------------------------------------------------------------------------
// Stage 3: softmax over 4096 logits (one 1024-thread block)
// ---------------------------------------------------------------------------
__global__ void __launch_bounds__(1024)
softmax4096_kernel(const float* __restrict__ lg, float* __restrict__ w_ws,
                   float* __restrict__ w_out) {
    __shared__ float sm[32];
    const int tid = threadIdx.x;
    float m = -INFINITY;
    for (int i = tid; i < 4096; i += blockDim.x) m = fmaxf(m, lg[i]);
    m = block_max(m, sm);
    float s = 0.0f;
    for (int i = tid; i < 4096; i += blockDim.x) s += expf(lg[i] - m);
    s = block_sum(s, sm);
    const float inv = 1.0f / s;
    for (int i = tid; i < 4096; i += blockDim.x) {
        const float v = expf(lg[i] - m) * inv;
        w_ws[i] = v;
        w_out[i] = v;
    }
}

// ---------------------------------------------------------------------------
// Stage 4: attn_applied = w^T @ enc  (split-K over 16 chunks, coalesced in j)
// ---------------------------------------------------------------------------
__global__ void __launch_bounds__(256)
attn_apply_partial_kernel(const float* __restrict__ w, const float* __restrict__ enc,
                          float* __restrict__ part) {
    const int p = blockIdx.x >> 2;                 // 16 row-chunks of 256
    const int j = (blockIdx.x & 3) * 256 + threadIdx.x;
    const int r0 = p * 256;
    float s = 0.0f;
    for (int r = r0; r < r0 + 256; ++r)
        s += w[r] * enc[(size_t)r * 1024 + j];     // coalesced across j
    part[p * 1024 + j] = s;
}
__global__ void __launch_bounds__(1024)
attn_apply_reduce_kernel(const float* __restrict__ part, float* __restrict__ outv) {
    const int j = threadIdx.x;
    float s = 0.0f;
#pragma unroll
    for (int p = 0; p < 16; ++p) s += part[p * 1024 + j];
    outv[j] = s;
}

// ---------------------------------------------------------------------------
// Stage 7: LSTM cell elementwise (gate order i,f,g,o)
// ---------------------------------------------------------------------------
__device__ inline float sigm(float x) { return 1.0f / (1.0f + expf(-x)); }

__global__ void __launch_bounds__(1024)
lstm_cell_kernel(const float* __restrict__ gates, const float* __restrict__ c0,
                 float* __restrict__ h1out, float* __restrict__ c1out,
                 float* __restrict__ h1ws) {
    const int j = threadIdx.x;
    const float ig = sigm(gates[j]);
    const float fg = sigm(gates[1024 + j]);
    const float gg = tanhf(gates[2048 + j]);
    const float og = sigm(gates[3072 + j]);
    const float c1 = fg * c0[j] + ig * gg;
    const float h1 = og * tanhf(c1);
    c1out[j] = c1;
    h1out[j] = h1;
    h1ws[j] = h1;
}

// ---------------------------------------------------------------------------
// Stage 9: log-softmax over 50257 (deterministic fixed-partition logsumexp)
// ---------------------------------------------------------------------------
__global__ void __launch_bounds__(256)
lse_partial_kernel(const float* __restrict__ lg, float* __restrict__ pm,
                   float* __restrict__ ps, int N) {
    __shared__ float sm[32];
    const int stride = gridDim.x * blockDim.x;
    const int g0 = blockIdx.x * blockDim.x + threadIdx.x;
    float m = -INFINITY;
    for (int i = g0; i < N; i += stride) m = fmaxf(m, lg[i]);
    m = block_max(m, sm);
    float s = 0.0f;
    for (int i = g0; i < N; i += stride) s += expf(lg[i] - m);
    s = block_sum(s, sm);
    if (threadIdx.x == 0) { pm[blockIdx.x] = m; ps[blockIdx.x] = s; }
}
__global__ void lse_final_kernel(const float* __restrict__ pm,
                                 const float* __restrict__ ps,
                                 float* __restrict__ lse, int P) {
    float M = -INFINITY;
    for (int p = 0; p < P; ++p) M = fmaxf(M, pm[p]);
    float S = 0.0f;
    for (int p = 0; p < P; ++p) S += ps[p] * expf(pm[p] - M);
    lse[0] = M + logf(S);
}
__global__ void __launch_bounds__(256)
write_logp_kernel(const float* __restrict__ lg, const float* __restrict__ lse,
                  float* __restrict__ out, int N) {
    const int i = blockIdx.x * blockDim.x + threadIdx.x;
    if (i < N) out[i] = lg[i] - lse[0];
}

// ---------------------------------------------------------------------------
// Host-side orchestration
// ---------------------------------------------------------------------------
extern "C" void kernel_launch(void* const* d_in, const int* in_sizes, int n_in,
                              void* d_out, int out_size, void* d_ws, size_t ws_size,
                              hipStream_t stream) {
    (void)in_sizes; (void)n_in; (void)out_size; (void)ws_size;
    constexpr int I = 1024, H = 1024, L = 4096, V = 50257;

    const int*   tok   = (const int*)d_in[0];
    const float* h0    = (const float*)d_in[1];
    const float* c0    = (const float*)d_in[2];
    const float* enc   = (const float*)d_in[3];
    const float* emb   = (const float*)d_in[4];
    const float* attnW = (const float*)d_in[5];
    const float* attnb = (const float*)d_in[6];
    const float* combW = (const float*)d_in[7];
    const float* combb = (const float*)d_in[8];
    const float* Wih   = (const float*)d_in[9];
    const float* bih   = (const float*)d_in[10];
    const float* Whh   = (const float*)d_in[11];
    const float* bhh   = (const float*)d_in[12];
    const float* outW  = (const float*)d_in[13];
    const float* outb  = (const float*)d_in[14];

    float* out = (float*)d_out;
    // d_out layout: logp[V] | h1[H] | c1[H] | attn_weights[L]
    float* out_logp = out;
    float* out_h1   = out + V;
    float* out_c1   = out + V + H;
    float* out_aw   = out + V + 2 * H;

    float* w = (float*)d_ws;  // ~341 KB of f32 scratch
    float* attn_in = w;            // 2048: [embedded | h0]
    float* vecB    = w + 2048;     // 2048: [embedded | attn_applied]
    float* alog    = w + 4096;     // 4096: attention logits
    float* aw      = w + 8192;     // 4096: attention weights
    float* part    = w + 12288;    // 16*1024 split-K partials
    float* xvec    = w + 28672;    // 1024: combine output
    float* gates   = w + 29696;    // 4096: LSTM gates
    float* h1ws    = w + 33792;    // 1024: h1 copy for projection input
    float* logits  = w + 34816;    // 50257: vocab logits
    float* pm      = w + 85248;    // 64 partial maxes
    float* ps      = w + 85312;    // 64 partial sums
    float* lse     = w + 85376;    // 1

    // 1) embedding gather + concat vectors
    build_vecs_kernel<<<8, 256, 0, stream>>>(tok, emb, h0, attn_in, vecB);
    // 2) attention logits: 4096x2048 GEMV (256 tiles, WMMA)
    wmma_gemv_kernel<<<32, 256, 0, stream>>>(attnW, attn_in, I + H,
                                             nullptr, nullptr, 0,
                                             attnb, nullptr, alog, L, 0);
    // 3) softmax over L
    softmax4096_kernel<<<1, 1024, 0, stream>>>(alog, aw, out_aw);
    // 4) attn_applied = w^T @ enc (split-K, then reduce)
    attn_apply_partial_kernel<<<64, 256, 0, stream>>>(aw, enc, part);
    attn_apply_reduce_kernel<<<1, 1024, 0, stream>>>(part, vecB + I);
    // 5) combine + ReLU: 1024x2048 GEMV (64 tiles, WMMA)
    wmma_gemv_kernel<<<8, 256, 0, stream>>>(combW, vecB, I + H,
                                            nullptr, nullptr, 0,
                                            combb, nullptr, xvec, I, 1);
    // 6) LSTM gates: fused W_ih@x + W_hh@h0 (256 tiles, WMMA)
    wmma_gemv_kernel<<<32, 256, 0, stream>>>(Wih, xvec, I, Whh, h0, H,
                                             bih, bhh, gates, 4 * H, 0);
    // 7) LSTM elementwise
    lstm_cell_kernel<<<1, 1024, 0, stream>>>(gates, c0, out_h1, out_c1, h1ws);
    // 8) vocab projection: 50257x1024 GEMV (3142 tiles -> 393 blocks, WMMA)
    wmma_gemv_kernel<<<393, 256, 0, stream>>>(outW, h1ws, H,
                                              nullptr, nullptr, 0,
                                              outb, nullptr, logits, V, 0);
    // 9) log-softmax over V
    lse_partial_kernel<<<64, 256, 0, stream>>>(logits, pm, ps, V);
    lse_final_kernel<<<1, 1, 0, stream>>>(pm, ps, lse, 64);
    write_logp_kernel<<<(V + 255) / 256, 256, 0, stream>>>(logits, lse, out_logp, V);
}